// Encoder_HieStackedCorr_52733608460806
// MI455X (gfx1250) — compile-verified
//
#include <hip/hip_runtime.h>
#include <math.h>
#include <stdint.h>

typedef __attribute__((ext_vector_type(2))) float v2f;
typedef __attribute__((ext_vector_type(8))) float v8f;

#define BB   32      // batch
#define NN   2048    // tokens
#define VD   256     // vdim
#define LRD  64      // low-rank dim
#define EMB  256     // embed

// workspace layout (float offsets)
#define WS_SCALE1 0                       // [64]
#define WS_SCALE2 64                      // [64]
#define WS_S      128                     // [32*64]   (zeroed, atomic-accumulated)
#define WS_VFINAL 2176                    // [32*256]  (zeroed, atomic-accumulated)
#define WS_DR     10368                   // [32*2048]
#define WS_COLSUM 75904                   // [32*2048]
#define WS_RIGHT  141440                  // [32*2048*64]

// ---------------- kernel 1: weight-norm scales + zero accumulators ----------
__global__ __launch_bounds__(256) void prep_kernel(const float* __restrict__ U1_v,
                                                   const float* __restrict__ U1_g,
                                                   const float* __restrict__ U2_v,
                                                   const float* __restrict__ U2_g,
                                                   float* __restrict__ ws) {
  int t = threadIdx.x;
  if (t < 64) {
    const float* r = U1_v + t * VD;
    float s = 0.f;
    for (int d = 0; d < VD; ++d) s += r[d] * r[d];
    ws[WS_SCALE1 + t] = U1_g[t] * rsqrtf(s);
  } else if (t < 128) {
    int l = t - 64;
    const float* r = U2_v + l * VD;
    float s = 0.f;
    for (int d = 0; d < VD; ++d) s += r[d] * r[d];
    ws[WS_SCALE2 + l] = U2_g[l] * rsqrtf(s);
  }
  // zero s[32*64] and v_final[32*256] (contiguous: WS_S .. WS_VFINAL+8192)
  for (int i = t; i < (BB * LRD + BB * VD); i += 256) ws[WS_S + i] = 0.f;
}

// ---- kernel 2: fused projections (WMMA f32) + diag/dr + s accumulation -----
// grid: (B*N)/16 blocks of 256 threads (8 waves).
// Waves 0-3: right = relu(V*W1^T + b1), cols 0-63 ; waves 4-7: left (W2).
__global__ __launch_bounds__(256) void proj_kernel(const float* __restrict__ Vmat,
                                                   const float* __restrict__ U1_v,
                                                   const float* __restrict__ U1_b,
                                                   const float* __restrict__ U2_v,
                                                   const float* __restrict__ U2_b,
                                                   float* __restrict__ ws) {
  __shared__ float ldsV[16 * 260];   // 16 tokens x 256, pad to kill bank conflicts
  __shared__ float act[16 * 128];    // [token][right(0..63) | left(64..127)]
  __shared__ float red[256];
  __shared__ float ldr[16];

  const int tid   = threadIdx.x;
  const int lane  = tid & 31;
  const int wave  = tid >> 5;
  const int token0 = blockIdx.x * 16;        // 16 tokens per tile, within one batch
  const int b      = token0 >> 11;           // N = 2048

  // ---- stage V tile [16 x 256] into LDS via async global->LDS DMA ----
  // (GLOBAL_LOAD_ASYNC_TO_LDS_B128: no VGPR round-trip, tracked by ASYNCcnt)
  {
    int row = tid >> 4;
    int col = (tid & 15) * 16;
    const float* g = Vmat + (size_t)token0 * VD + (size_t)row * VD + col;
    unsigned lds_base = (unsigned)(uintptr_t)(&ldsV[row * 260 + col]);
#pragma unroll
    for (int i = 0; i < 4; ++i) {
      unsigned lofs = lds_base + i * 16;      // LDS byte address (low 32 aperture bits)
      const float* ga = g + i * 4;            // 16B per lane
      asm volatile("global_load_async_to_lds_b128 %0, %1, off"
                   :: "v"(lofs), "v"(ga)
                   : "memory");
    }
    asm volatile("s_wait_asynccnt 0x0" ::: "memory");
  }
  __syncthreads();

  const int sel    = wave >> 2;            // 0 = right(U1), 1 = left(U2)
  const int colblk = (wave & 3) * 16;
  const int n16    = lane & 15;
  const int koff   = (lane < 16) ? 0 : 2;  // K sub-pair per half-wave (ISA layout)
  const int l      = colblk + n16;         // output column (LR index)
  const float* urow = (sel ? U2_v : U1_v) + (size_t)l * VD;  // B[K][N]=U[l][k]
  const float* arow = ldsV + (lane & 15) * 260;              // A row = token M

  v8f acc = {};
#pragma unroll 8
  for (int kk = 0; kk < 64; ++kk) {
    int k = kk * 4 + koff;
    v2f a  = *(const v2f*)(arow + k);   // A[M][k], A[M][k+1]
    v2f bf = *(const v2f*)(urow + k);   // B[k][N], B[k+1][N]
    acc = __builtin_amdgcn_wmma_f32_16x16x4_f32(false, a, false, bf,
                                                (short)0, acc, false, false);
  }

  // fused epilogue: weight-norm scale + bias + ReLU; stash to LDS (+global for right)
  const float sc   = sel ? ws[WS_SCALE2 + l] : ws[WS_SCALE1 + l];
  const float bias = sel ? U2_b[l] : U1_b[l];
  float* rightg = ws + WS_RIGHT;
#pragma unroll
  for (int r = 0; r < 8; ++r) {
    int token = r + ((lane >> 4) << 3);   // C/D layout: VGPR r -> M=r (+8 upper half)
    float v = fmaxf(acc[r] * sc + bias, 0.f);
    act[token * 128 + sel * 64 + l] = v;
    if (sel == 0) rightg[(size_t)(token0 + token) * LRD + l] = v;
  }
  __syncthreads();

  // diag d[token] = sum_l left*right ; dr = rsqrt(d + eps)
  {
    int token = tid & 15, grp = tid >> 4;
    float p = 0.f;
#pragma unroll
    for (int j = 0; j < 4; ++j) {
      int ll = grp * 4 + j;
      p += act[token * 128 + ll] * act[token * 128 + 64 + ll];
    }
    red[grp * 16 + token] = p;
  }
  __syncthreads();
  if (tid < 16) {
    float dsum = 0.f;
#pragma unroll
    for (int g = 0; g < 16; ++g) dsum += red[g * 16 + tid];
    float drv = rsqrtf(dsum + 1e-6f);
    ldr[tid] = drv;
    ws[WS_DR + token0 + tid] = drv;
  }
  __syncthreads();

  // s[b,l] += sum_token dr[token] * left[token,l]
  if (tid < 64) {
    float ps = 0.f;
#pragma unroll
    for (int t2 = 0; t2 < 16; ++t2) ps += ldr[t2] * act[t2 * 128 + 64 + tid];
    atomicAdd(&ws[WS_S + b * LRD + tid], ps);
  }
}

// ------ kernel 3: colsum[b,m] = (N+1) - dr[b,m] * dot(s[b,:], right[b,m,:]) ----
__global__ __launch_bounds__(256) void colsum_kernel(float* __restrict__ ws) {
  __shared__ float sl[64];
  int gm = blockIdx.x * 256 + threadIdx.x;   // flat (b,m); block stays in one batch
  int b  = gm >> 11;
  if (threadIdx.x < 64) sl[threadIdx.x] = ws[WS_S + b * LRD + threadIdx.x];
  __syncthreads();
  const float4* r = (const float4*)(ws + WS_RIGHT + (size_t)gm * LRD);
  float dot = 0.f;
#pragma unroll
  for (int i = 0; i < 16; ++i) {
    float4 v = r[i];
    dot += v.x * sl[i * 4] + v.y * sl[i * 4 + 1] + v.z * sl[i * 4 + 2] + v.w * sl[i * 4 + 3];
  }
  ws[WS_COLSUM + gm] = (float)(NN + 1) - ws[WS_DR + gm] * dot;
}

// ------ kernel 4: v_final[b,:] += (1/N) * sum_m colsum[b,m] * Vmat[b,m,:] ------
__global__ __launch_bounds__(256) void vfinal_kernel(const float* __restrict__ Vmat,
                                                     float* __restrict__ ws) {
  int b  = blockIdx.y;
  int m0 = blockIdx.x * 128;
  int d  = threadIdx.x;
  const float* vb = Vmat + ((size_t)b * NN + m0) * VD;
  const float* cs = ws + WS_COLSUM + b * NN + m0;
  float acc = 0.f;
#pragma unroll 4
  for (int mm = 0; mm < 128; ++mm) acc += cs[mm] * vb[(size_t)mm * VD + d];
  atomicAdd(&ws[WS_VFINAL + b * VD + d], acc * (1.0f / NN));
}

// ------ kernel 5: feat = v_final @ W_lin^T + b ; BatchNorm(train) over B ------
__global__ __launch_bounds__(256) void head_kernel(const float* __restrict__ W_lin,
                                                   const float* __restrict__ b_lin,
                                                   const float* __restrict__ gamma,
                                                   const float* __restrict__ beta,
                                                   const float* __restrict__ ws,
                                                   float* __restrict__ out) {
  __shared__ float vf[BB * VD];
  int t = threadIdx.x;                       // t = embed channel e
  for (int i = t; i < BB * VD; i += 256) vf[i] = ws[WS_VFINAL + i];
  __syncthreads();
  const float* wrow = W_lin + (size_t)t * VD;
  float f[BB];
  float mu = 0.f;
  for (int bb = 0; bb < BB; ++bb) {
    float s = 0.f;
    for (int d = 0; d < VD; ++d) s += vf[bb * VD + d] * wrow[d];
    f[bb] = s + b_lin[t];
    mu += f[bb];
  }
  mu *= (1.0f / BB);
  float var = 0.f;
  for (int bb = 0; bb < BB; ++bb) { float dd = f[bb] - mu; var += dd * dd; }
  var *= (1.0f / BB);
  float inv = rsqrtf(var + 1e-5f) * gamma[t];
  for (int bb = 0; bb < BB; ++bb) out[bb * EMB + t] = (f[bb] - mu) * inv + beta[t];
}

extern "C" void kernel_launch(void* const* d_in, const int* in_sizes, int n_in,
                              void* d_out, int out_size, void* d_ws, size_t ws_size,
                              hipStream_t stream) {
  (void)in_sizes; (void)n_in; (void)out_size; (void)ws_size;
  const float* Vmat  = (const float*)d_in[0];
  const float* U1_v  = (const float*)d_in[1];
  const float* U1_g  = (const float*)d_in[2];
  const float* U1_b  = (const float*)d_in[3];
  const float* U2_v  = (const float*)d_in[4];
  const float* U2_g  = (const float*)d_in[5];
  const float* U2_b  = (const float*)d_in[6];
  const float* W_lin = (const float*)d_in[7];
  const float* b_lin = (const float*)d_in[8];
  const float* gamma = (const float*)d_in[9];
  const float* beta  = (const float*)d_in[10];
  float* out = (float*)d_out;
  float* ws  = (float*)d_ws;

  prep_kernel<<<1, 256, 0, stream>>>(U1_v, U1_g, U2_v, U2_g, ws);
  proj_kernel<<<(BB * NN) / 16, 256, 0, stream>>>(Vmat, U1_v, U1_b, U2_v, U2_b, ws);
  colsum_kernel<<<(BB * NN) / 256, 256, 0, stream>>>(ws);
  vfinal_kernel<<<dim3(16, BB), 256, 0, stream>>>(Vmat, ws);
  head_kernel<<<1, 256, 0, stream>>>(W_lin, b_lin, gamma, beta, ws, out);
}